// KernelConv_81312320848612
// MI455X (gfx1250) — compile-verified
//
#include <hip/hip_runtime.h>

// Problem constants (from reference): data [4,64,128,128] f32, core [4,25*64,128,128] f32
constexpr int KB   = 4;     // batch
constexpr int KC   = 64;    // channels
constexpr int KH   = 128;
constexpr int KW   = 128;
constexpr int KK   = 5;     // kernel size
constexpr int KP   = 2;     // pad
constexpr int TAPS = KK * KK;  // 25

constexpr int ROWS  = 8;                // output rows per block
constexpr int TW    = KW + 2 * KP;      // 132 tile width (floats), 16B-aligned rows (132 = 4*33)
constexpr int THH   = ROWS + 2 * KP;    // 12 tile height
constexpr int TILE_ELEMS = THH * TW;    // 1584 floats = 6336 B LDS
constexpr int BLOCK = 256;              // 8 waves (wave32)

typedef float v4f __attribute__((ext_vector_type(4)));

// gfx1250 async global->LDS DMA (ASYNCcnt path). GV mode: 64-bit per-lane address.
__device__ __forceinline__ void async_load_f32_to_lds(unsigned lds_byte_off,
                                                      const float* gptr) {
  asm volatile("global_load_async_to_lds_b32 %0, %1, off"
               :: "v"(lds_byte_off), "v"((unsigned long long)gptr)
               : "memory");
}

__device__ __forceinline__ void wait_asynccnt0() {
  asm volatile("s_wait_asynccnt 0x0" ::: "memory");
}

__global__ __launch_bounds__(BLOCK)
void kpn_dynconv_kernel(const float* __restrict__ data,
                        const float* __restrict__ core,
                        float* __restrict__ out) {
  __shared__ __align__(16) float tile[TILE_ELEMS];

  const unsigned tid = threadIdx.x;
  const unsigned bid = blockIdx.x;
  constexpr unsigned NHB = KH / ROWS;      // 16 row-strips per plane
  const unsigned hb = bid % NHB;
  const unsigned bc = bid / NHB;
  const unsigned c  = bc % KC;
  const unsigned b  = bc / KC;
  const unsigned h0 = hb * ROWS;

  // ---- Stage data tile (12 x 132, zero halo) into LDS via async global->LDS DMA
  const float* dplane = data + (size_t)(b * KC + c) * (KH * KW);
  const unsigned ldsBase = (unsigned)(unsigned long long)(&tile[0]);

  for (unsigned idx = tid; idx < (unsigned)TILE_ELEMS; idx += BLOCK) {
    const unsigned tr  = idx / TW;
    const unsigned col = idx - tr * TW;
    const int gh = (int)(h0 + tr) - KP;
    const int gw = (int)col - KP;
    if ((unsigned)gh < (unsigned)KH && (unsigned)gw < (unsigned)KW) {
      async_load_f32_to_lds(ldsBase + idx * 4u, dplane + gh * KW + gw);
    } else {
      tile[idx] = 0.0f;  // zero padding halo (disjoint from async targets)
    }
  }
  wait_asynccnt0();    // own wave's async DMAs landed in LDS
  __syncthreads();     // + all waves' DMAs and zero-fill stores visible

  // ---- Each thread: 4 consecutive output pixels along W
  const unsigned r    = tid >> 5;       // 0..7 output row within strip
  const unsigned lane = tid & 31u;
  const unsigned w0   = lane * 4u;
  const unsigned h    = h0 + r;

  // 5 x 8 register window from LDS (10x ds_load_b128, 16B aligned)
  float d[KK][8];
#pragma unroll
  for (int i = 0; i < KK; ++i) {
    const v4f lo = *(const v4f*)(&tile[(r + i) * TW + w0]);
    const v4f hi = *(const v4f*)(&tile[(r + i) * TW + w0 + 4]);
    d[i][0] = lo[0]; d[i][1] = lo[1]; d[i][2] = lo[2]; d[i][3] = lo[3];
    d[i][4] = hi[0]; d[i][5] = hi[1]; d[i][6] = hi[2]; d[i][7] = hi[3];
  }

  // 25 streaming non-temporal b128 loads of the per-pixel kernels (419 MB stream;
  // NT keeps the L2-resident data tensor from being evicted). Issue all loads
  // up front for latency hiding, then do the 100 FMAs.
  const size_t plane = (size_t)KC * KH * KW;  // stride between taps (1,048,576 floats)
  const float* cbase = core + (((size_t)b * TAPS * KC + c) * KH + h) * KW + w0;

  v4f cv[TAPS];
#pragma unroll
  for (int t = 0; t < TAPS; ++t) {
    cv[t] = __builtin_nontemporal_load((const v4f*)(cbase + (size_t)t * plane));
  }

  v4f acc = {0.f, 0.f, 0.f, 0.f};
#pragma unroll
  for (int t = 0; t < TAPS; ++t) {
    const int i = t / KK;
    const int j = t % KK;
    acc[0] += cv[t][0] * d[i][j + 0];
    acc[1] += cv[t][1] * d[i][j + 1];
    acc[2] += cv[t][2] * d[i][j + 2];
    acc[3] += cv[t][3] * d[i][j + 3];
  }

  float* optr = out + ((size_t)(b * KC + c) * KH + h) * KW + w0;
  __builtin_nontemporal_store(acc, (v4f*)optr);
}

extern "C" void kernel_launch(void* const* d_in, const int* in_sizes, int n_in,
                              void* d_out, int out_size, void* d_ws, size_t ws_size,
                              hipStream_t stream) {
  const float* data = (const float*)d_in[0];  // [4,64,128,128] f32
  const float* core = (const float*)d_in[1];  // [4,1600,128,128] f32
  float* out = (float*)d_out;                 // [4,64,128,128] f32
  const int grid = KB * KC * (KH / ROWS);     // 4096 blocks x 256 threads
  hipLaunchKernelGGL(kpn_dynconv_kernel, dim3(grid), dim3(BLOCK), 0, stream,
                     data, core, out);
}